// H_50259707298064
// MI455X (gfx1250) — compile-verified
//
#include <hip/hip_runtime.h>

// ---------------------------------------------------------------------------
// Hadamard on wires {0,5,9} of a 13-qubit state [8192, 2048] complex64.
// DIM=2 => H is real: radix-8 sign butterfly (scale 2^-1.5) over row bits
// {12,7,3}, applied independently to the real and imag planes.
//
// Memory-bound: 256 MB traffic -> ~11 us at 23.3 TB/s. Each input element is
// read exactly once. The 8-point butterfly is a 16x16 block-diagonal fp32
// GEMM (two butterfly groups per wave) executed as 4 chained
// V_WMMA_F32_16X16X4_F32 per plane (full fp32 precision).
//
// v2: amortize the per-wave setup (A matrix, row/store offsets) over 8 column
// tiles per wave to keep instruction-issue off the critical path, and use
// non-temporal loads/stores since this is a pure streaming workload
// (256 MB working set > 192 MB L2, every byte touched once).
// ---------------------------------------------------------------------------

typedef float v2f __attribute__((ext_vector_type(2)));
typedef float v8f __attribute__((ext_vector_type(8)));

constexpr int   BATCH  = 2048;
constexpr int   BIT0   = 3;      // wire 9  -> bit 3
constexpr int   BIT1   = 7;      // wire 5  -> bit 7
constexpr int   BIT2   = 12;     // wire 0  -> bit 12
constexpr float HSCALE = 0.35355339059327379f;  // 2^(-3/2)

// Spread a 10-bit group id into row bits, leaving holes at bits 3, 7, 12.
__device__ __forceinline__ int spread_group(int g) {
  int low3 = g & 7;          // -> bits 0..2
  int mid3 = (g >> 3) & 7;   // -> bits 4..6
  int hi4  = (g >> 6) & 15;  // -> bits 8..11
  return low3 | (mid3 << 4) | (hi4 << 8);
}

// Row offset of butterfly element j (j in [0,8)) within a group.
__device__ __forceinline__ int rowoff(int j) {
  return ((j & 1) << BIT0) | (((j >> 1) & 1) << BIT1) | (((j >> 2) & 1) << BIT2);
}

// T16[m][k]: block-diagonal pair of 8x8 sign-butterfly matrices.
__device__ __forceinline__ float tval(int m, int k) {
  if ((m ^ k) & 8) return 0.0f;                 // off-diagonal block
  return (__popc(m & k & 7) & 1) ? -HSCALE : HSCALE;
}

__global__ __launch_bounds__(256) void hadamard_wmma_kernel(
    const float* __restrict__ xr,   // [8192, 2048] real plane
    const float* __restrict__ xi,   // [8192, 2048] imag plane
    v2f* __restrict__ out)          // [8192, 2048] complex64 (re, im)
{
  const int lane = threadIdx.x & 31;
  const int wave = threadIdx.x >> 5;
  const int gPair    = blockIdx.x >> 1;    // 512 group pairs
  const int colHalf  = blockIdx.x & 1;     // 2 halves of the 128 column tiles

  const int N  = lane & 15;                // matrix column / A-row within lane
  const int hi = lane >> 4;                // upper lane half selector

  const int rb0 = spread_group(2 * gPair);      // base row, butterfly group 0
  const int rb1 = spread_group(2 * gPair + 1);  // base row, butterfly group 1

  // --- A matrix: constant transform T16, per documented 16x4 f32 layout:
  //     lane l holds M = l&15; VGPR0 = K=(2*hi), VGPR1 = K=(2*hi+1) of chunk.
  //     Invariant across all column tiles -> hoisted out of the loop.
  v2f a[4];
#pragma unroll
  for (int s = 0; s < 4; ++s) {
    a[s].x = tval(N, 4 * s + 2 * hi);
    a[s].y = tval(N, 4 * s + 2 * hi + 1);
  }

  // --- Per-lane element offsets (identical for xr and xi), invariant across
  //     column tiles. Kept 32-bit so the backend can use saddr+voffset form.
  unsigned off0[4], off1[4];
#pragma unroll
  for (int s = 0; s < 4; ++s) {
    const int k  = 4 * s + 2 * hi;          // global input-row index 0..15
    const int rb = (k < 8) ? rb0 : rb1;     // which butterfly group
    const int r0 = rb + rowoff(k & 7);      // even j; odd j row = r0 + (1<<BIT0)
    off0[s] = (unsigned)r0 * BATCH + N;
    off1[s] = off0[s] + 8u * BATCH;         // +8 rows (bit 3 set)
  }

  unsigned sto[8];
  const int rbs = hi ? rb1 : rb0;           // lane's output group
#pragma unroll
  for (int v = 0; v < 8; ++v)
    sto[v] = (unsigned)(rbs + rowoff(v)) * BATCH + N;

  // --- 8 column tiles per wave: 1024 blocks * 8 waves * 8 iters = 65536 tiles
  for (int t = 0; t < 8; ++t) {
    const unsigned colBase = (unsigned)(colHalf * 64 + t * 8 + wave) * 16;

    // B matrix: 16 input rows x 16 batch columns, in 4 K-chunks of 4.
    // lane l holds column N; VGPR0/1 = rows K = 2*hi, 2*hi+1 of the chunk.
    v2f br[4], bim[4];
#pragma unroll
    for (int s = 0; s < 4; ++s) {
      const unsigned i0 = off0[s] + colBase;
      const unsigned i1 = off1[s] + colBase;
      br[s].x  = __builtin_nontemporal_load(xr + i0);
      br[s].y  = __builtin_nontemporal_load(xr + i1);
      bim[s].x = __builtin_nontemporal_load(xi + i0);
      bim[s].y = __builtin_nontemporal_load(xi + i1);
    }

    // D = T16 * X, fp32 accumulate over K = 16 (4 chained WMMAs per plane).
    v8f cr = {};
    v8f ci = {};
#pragma unroll
    for (int s = 0; s < 4; ++s) {
      cr = __builtin_amdgcn_wmma_f32_16x16x4_f32(false, a[s], false, br[s],
                                                 (short)0, cr, false, false);
      ci = __builtin_amdgcn_wmma_f32_16x16x4_f32(false, a[s], false, bim[s],
                                                 (short)0, ci, false, false);
    }

    // Store: C/D layout = VGPR v, lanes 0-15 -> M=v, lanes 16-31 -> M=v+8.
    // Interleaved complex64 as 8-byte vector stores, non-temporal (streamed).
#pragma unroll
    for (int v = 0; v < 8; ++v) {
      v2f val;
      val.x = cr[v];
      val.y = ci[v];
      __builtin_nontemporal_store(val, out + (sto[v] + colBase));
    }
  }
}

extern "C" void kernel_launch(void* const* d_in, const int* in_sizes, int n_in,
                              void* d_out, int out_size, void* d_ws, size_t ws_size,
                              hipStream_t stream) {
  const float* xr = (const float*)d_in[0];
  const float* xi = (const float*)d_in[1];
  v2f* out = (v2f*)d_out;

  // 512 group pairs * 2 column halves = 1024 blocks of 8 waves;
  // each wave processes 8 (groupPair, 16-column) tiles.
  dim3 grid(1024), block(256);
  hipLaunchKernelGGL(hadamard_wmma_kernel, grid, block, 0, stream, xr, xi, out);
}